// RelPGExplainer_57123065036979
// MI455X (gfx1250) — compile-verified
//
#include <hip/hip_runtime.h>
#include <math.h>

typedef __attribute__((ext_vector_type(2))) float v2f;
typedef __attribute__((ext_vector_type(8))) float v8f;

#define GBIAS 0.0001f
#define TEMP_INV 0.2f

// ---------------------------------------------------------------------------
// Kernel 1: per-batch constant
//   cb[b] = dot(node[h_index[b]], W[384:512]) + dot(R[r_index[b]], W[512:640]) + bias
// One wave32 per batch element; 4 floats per lane; shuffle reduction.
// ---------------------------------------------------------------------------
__global__ void relpg_batch_term(const float* __restrict__ node,
                                 const float* __restrict__ Rm,
                                 const float* __restrict__ W,
                                 const float* __restrict__ bias,
                                 const int* __restrict__ h_index,
                                 const int* __restrict__ r_index,
                                 float* __restrict__ cb, int Bn) {
    int wave = (blockIdx.x * blockDim.x + threadIdx.x) >> 5;
    int lane = threadIdx.x & 31;
    if (wave >= Bn) return;
    const float4 h4 = *(const float4*)(node + (size_t)h_index[wave] * 128 + lane * 4);
    const float4 r4 = *(const float4*)(Rm + (size_t)r_index[wave] * 128 + lane * 4);
    const float4 wh = *(const float4*)(W + 384 + lane * 4);
    const float4 wr = *(const float4*)(W + 512 + lane * 4);
    float p = h4.x * wh.x + h4.y * wh.y + h4.z * wh.z + h4.w * wh.w +
              r4.x * wr.x + r4.y * wr.y + r4.z * wr.z + r4.w * wr.w;
    for (int off = 16; off > 0; off >>= 1) p += __shfl_xor(p, off, 32);
    if (lane == 0) cb[wave] = p + bias[0];
}

// ---------------------------------------------------------------------------
// Kernel 2: fused gather + GEMV (via v_wmma_f32_16x16x4_f32) + gumbel-sigmoid.
// One wave handles 16 edges. A: 16 gathered rows x 4 K-slice (fp32).
// B: W slice replicated over N (so every D column equals the dot result).
// ---------------------------------------------------------------------------
__global__ void relpg_edge_gate(const float* __restrict__ node,
                                const float* __restrict__ Rm,
                                const float* __restrict__ W,
                                const float* __restrict__ eps_u,
                                const int* __restrict__ rows,
                                const int* __restrict__ cols,
                                const int* __restrict__ etype,
                                const int* __restrict__ batch_id,
                                const float* __restrict__ cb,
                                float* __restrict__ out, int E) {
    __shared__ float sW[384];
    if (threadIdx.x < 96)
        *(float4*)(sW + threadIdx.x * 4) = *(const float4*)(W + threadIdx.x * 4);
    __syncthreads();

    const int lane = threadIdx.x & 31;
    const int wave = threadIdx.x >> 5;
    const int m    = lane & 15;     // edge-in-group (A-matrix row M)
    const int kh   = lane >> 4;     // lane half selects K pair
    const int c0   = kh * 2;
    const int e0   = blockIdx.x * 128 + wave * 16;

    int em = e0 + m;
    if (em >= E) em = E - 1;        // clamp so EXEC stays all-ones for WMMA
    const float* pr = node + (size_t)rows[em]  * 128;
    const float* pc = node + (size_t)cols[em]  * 128;
    const float* pt = Rm   + (size_t)etype[em] * 128;

    v8f acc0 = {0.f, 0.f, 0.f, 0.f, 0.f, 0.f, 0.f, 0.f};
    v8f acc1 = acc0;
    v8f acc2 = acc0;

    #pragma unroll 4
    for (int kk = 0; kk < 128; kk += 4) {
        const int o = kk + c0;
        v2f a0 = *(const v2f*)(pr + o);
        v2f a1 = *(const v2f*)(pc + o);
        v2f a2 = *(const v2f*)(pt + o);
        v2f b0 = *(const v2f*)(sW + o);
        v2f b1 = *(const v2f*)(sW + 128 + o);
        v2f b2 = *(const v2f*)(sW + 256 + o);
        acc0 = __builtin_amdgcn_wmma_f32_16x16x4_f32(false, a0, false, b0,
                                                     (short)0, acc0, false, false);
        acc1 = __builtin_amdgcn_wmma_f32_16x16x4_f32(false, a1, false, b1,
                                                     (short)0, acc1, false, false);
        acc2 = __builtin_amdgcn_wmma_f32_16x16x4_f32(false, a2, false, b2,
                                                     (short)0, acc2, false, false);
    }

    // D layout: VGPR v, lanes 0-15 -> M=v; lanes 16-31 -> M=v+8 (all N identical).
    // Lane 0 emits edges e0..e0+7, lane 16 emits e0+8..e0+15.
    if (m == 0) {
        #pragma unroll
        for (int v = 0; v < 8; ++v) {
            int e = e0 + kh * 8 + v;
            if (e < E) {
                float w   = acc0[v] + acc1[v] + acc2[v] + cb[batch_id[e]];
                float u   = eps_u[e];
                float eps = (2.f * GBIAS - 1.f) * u + (1.f - GBIAS);
                float gate = (logf(eps) - log1pf(-eps) + w) * TEMP_INV;
                out[e] = 1.f / (1.f + expf(-gate));
            }
        }
    }
}

// ---------------------------------------------------------------------------
extern "C" void kernel_launch(void* const* d_in, const int* in_sizes, int n_in,
                              void* d_out, int out_size, void* d_ws, size_t ws_size,
                              hipStream_t stream) {
    const float* node   = (const float*)d_in[0];   // (N,128)
    const float* Rm     = (const float*)d_in[1];   // (NUM_REL,128)
    const float* W      = (const float*)d_in[2];   // (640,1)
    const float* bias   = (const float*)d_in[3];   // (1,)
    const float* eps_u  = (const float*)d_in[4];   // (E,1)
    const int*   eidx   = (const int*)d_in[5];     // (2,E)
    const int*   etype  = (const int*)d_in[6];     // (E,)
    const int*   bid    = (const int*)d_in[7];     // (E,)
    const int*   h_idx  = (const int*)d_in[8];     // (B,)
    const int*   r_idx  = (const int*)d_in[9];     // (B,)

    const int E  = in_sizes[4];
    const int Bn = in_sizes[8];
    const int* rows = eidx;
    const int* cols = eidx + E;

    float* cb  = (float*)d_ws;                     // Bn floats of scratch
    float* out = (float*)d_out;

    // Kernel 1: 8 waves per block, one wave per batch element.
    int blocks1 = (Bn * 32 + 255) / 256;
    relpg_batch_term<<<blocks1, 256, 0, stream>>>(node, Rm, W, bias, h_idx, r_idx, cb, Bn);

    // Kernel 2: 128 edges per 256-thread block (16 edges per wave).
    int blocks2 = (E + 127) / 128;
    relpg_edge_gate<<<blocks2, 256, 0, stream>>>(node, Rm, W, eps_u, rows, cols,
                                                 etype, bid, cb, out, E);
}